// SimilarityUpsamplerBatch_15075335208982
// MI455X (gfx1250) — compile-verified
//
#include <hip/hip_runtime.h>
#include <stdint.h>

// Reference shapes: B=16, T_RED=512, C=512. max_len is recovered from out_size.
namespace {
constexpr int kB    = 16;
constexpr int kTred = 512;
constexpr int kC    = 512;
}

// --------------------------------------------------------------------------
// Kernel 1: per-batch stable partition (positives first) + inclusive scan.
// One block per batch, 512 threads (16 wave32s). Two LDS Hillis-Steele scans.
// Also computes curr_len[b] = min(total, target_T[b]).
// --------------------------------------------------------------------------
__global__ void __launch_bounds__(kTred)
durscan_kernel(const int* __restrict__ durations,
               const int* __restrict__ target_T,
               int* __restrict__ csum,       // [B, T_RED]
               int* __restrict__ curr_len) { // [B]
  __shared__ int s[kTred];
  __shared__ int part[kTred];
  const int b = blockIdx.x;
  const int i = threadIdx.x;

  const int d     = durations[b * kTred + i];
  const int isPos = (d > 0) ? 1 : 0;

  // Inclusive scan of the positive-indicator -> stable partition ranks.
  s[i] = isPos;
  __syncthreads();
  for (int off = 1; off < kTred; off <<= 1) {
    const int t = (i >= off) ? s[i - off] : 0;
    __syncthreads();
    s[i] += t;
    __syncthreads();
  }
  const int posRank = s[i];            // inclusive count of positives <= i
  const int numPos  = s[kTred - 1];
  const int dest    = isPos ? (posRank - 1)
                            : (numPos + (i + 1 - posRank) - 1);
  part[dest] = d;
  __syncthreads();

  // Inclusive scan of the partitioned durations -> cumsum.
  s[i] = part[i];
  __syncthreads();
  for (int off = 1; off < kTred; off <<= 1) {
    const int t = (i >= off) ? s[i - off] : 0;
    __syncthreads();
    s[i] += t;
    __syncthreads();
  }
  csum[b * kTred + i] = s[i];
  if (i == 0) {
    const int total = s[kTred - 1];
    const int tt    = target_T[b];
    curr_len[b] = (total < tt) ? total : tt;
  }
}

// --------------------------------------------------------------------------
// Kernel 2: upsample via gather. Grid (ceil(max_len/2), B), 256 threads.
// Each 128-thread half-block owns one output frame t:
//   - branch-free binary search of csum (searchsorted-right), clamp to 511
//   - copy 512 floats via CDNA5 async DMA: global -> LDS -> global
//     (GLOBAL_LOAD_ASYNC_TO_LDS_B128 / GLOBAL_STORE_ASYNC_FROM_LDS_B128,
//      synchronized with s_wait_asynccnt; each lane moves its own 16B so no
//      cross-lane LDS sharing and no workgroup barrier is needed)
//   - the LDS->global store uses TH_STORE_NT: the 131 MB write-once output
//     stream must not evict the 16 MB, ~8x-reused gather source from the
//     192 MB L2 (keeps HBM reads near the 16 MB minimum)
//   - invalid frames store zeros; mask written as 0.0f / 1.0f (both NT)
// --------------------------------------------------------------------------
__global__ void __launch_bounds__(256)
upsample_kernel(const float* __restrict__ x,        // [B, T_RED, C]
                const int*   __restrict__ csum,     // [B, T_RED]
                const int*   __restrict__ curr_len, // [B]
                float*       __restrict__ out,      // [B, max_len, C]
                float*       __restrict__ mask,     // [B, max_len]
                int max_len) {
#if defined(__gfx1250__)
  __shared__ float smem[2][kC];   // 4 KB staging, one row per half-block
#endif
  const int b    = blockIdx.y;
  const int half = threadIdx.x >> 7;   // 0 or 1: which frame in this block
  const int lane = threadIdx.x & 127;  // 0..127: 4 floats (16B) per lane
  const int t    = blockIdx.x * 2 + half;
  if (t >= max_len) return;            // no barriers below -> safe early exit

  const int* __restrict__ cs = csum + b * kTred;
  if (threadIdx.x == 0) __builtin_prefetch(cs, 0, 0);  // global_prefetch_b8

  // searchsorted(cs, t, side='right'): first index with cs[idx] > t.
  int lo = 0, hi = kTred;
#pragma unroll
  for (int it = 0; it < 9; ++it) {     // log2(512) = 9 steps, branch-free
    const int  mid = (lo + hi) >> 1;
    const bool le  = (cs[mid] <= t);
    lo = le ? (mid + 1) : lo;
    hi = le ? hi : mid;
  }
  const int  seg   = (lo < kTred - 1) ? lo : (kTred - 1);
  const bool valid = (t < curr_len[b]);

  float* __restrict__ dst = out + ((size_t)b * max_len + t) * kC + lane * 4;

  if (valid) {
    const float* __restrict__ src =
        x + ((size_t)(b * kTred + seg)) * kC + lane * 4;
#if defined(__gfx1250__)
    // Low 32 bits of a generic LDS pointer are the LDS byte offset.
    const uint32_t lofs = (uint32_t)(uintptr_t)(&smem[half][lane * 4]);
    asm volatile("global_load_async_to_lds_b128 %0, %1, off"
                 :: "v"(lofs), "v"(src) : "memory");
    asm volatile("s_wait_asynccnt 0x0" ::: "memory");
    asm volatile("global_store_async_from_lds_b128 %0, %1, off th:TH_STORE_NT"
                 :: "v"(dst), "v"(lofs) : "memory");
    // S_ENDPGM performs an implicit wait-idle, draining ASYNCcnt.
#else
    *(float4*)dst = *(const float4*)src;   // host-pass / fallback path
#endif
  } else {
    // Non-temporal zero fill: write-once data, keep it out of L2.
    __builtin_nontemporal_store(0.f, dst + 0);
    __builtin_nontemporal_store(0.f, dst + 1);
    __builtin_nontemporal_store(0.f, dst + 2);
    __builtin_nontemporal_store(0.f, dst + 3);
  }
  if (lane == 0) {
    __builtin_nontemporal_store(valid ? 0.0f : 1.0f,
                                mask + (size_t)b * max_len + t);
  }
}

// --------------------------------------------------------------------------
// Host launcher.
// d_in[0] = pooled_x  (float32, B*T_RED*C)
// d_in[1] = durations (int32,  B*T_RED)
// d_in[2] = target_T  (int32,  B)
// d_out   = x_restored (B*max_len*C f32) ++ new_mask (B*max_len, stored as f32)
// --------------------------------------------------------------------------
extern "C" void kernel_launch(void* const* d_in, const int* in_sizes, int n_in,
                              void* d_out, int out_size, void* d_ws, size_t ws_size,
                              hipStream_t stream) {
  (void)in_sizes; (void)n_in; (void)ws_size;
  const float* pooled_x  = (const float*)d_in[0];
  const int*   durations = (const int*)d_in[1];
  const int*   target_T  = (const int*)d_in[2];

  const int max_len = out_size / (kB * (kC + 1));

  int* csum     = (int*)d_ws;            // 16*512*4 = 32 KB
  int* curr_len = csum + kB * kTred;     // +64 B

  durscan_kernel<<<kB, kTred, 0, stream>>>(durations, target_T, csum, curr_len);

  if (max_len > 0) {
    float* out_x    = (float*)d_out;
    float* out_mask = out_x + (size_t)kB * max_len * kC;
    dim3 grid((max_len + 1) / 2, kB);
    upsample_kernel<<<grid, 256, 0, stream>>>(pooled_x, csum, curr_len,
                                              out_x, out_mask, max_len);
  }
}